// SparseReservoir_1245540516174
// MI455X (gfx1250) — compile-verified
//
#include <hip/hip_runtime.h>
#include <stdint.h>

#define AS1 __attribute__((address_space(1)))
#define AS3 __attribute__((address_space(3)))

typedef __attribute__((ext_vector_type(2))) float v2f;
typedef __attribute__((ext_vector_type(8))) float v8f;

// ---- CDNA5 async global->LDS copy (ASYNCcnt path), with safe fallback ----
__device__ __forceinline__ void async_g2l_b32(const void* gsrc, void* ldst) {
#if __has_builtin(__builtin_amdgcn_global_load_async_to_lds_b32)
  void* gp = (void*)gsrc;  // strip const, then addrspace-cast to typed pointers
  __builtin_amdgcn_global_load_async_to_lds_b32((AS1 int*)gp, (AS3 int*)ldst,
                                                /*offset=*/0, /*cpol=*/0);
#else
  *(int*)ldst = *(const int*)gsrc;
#endif
}

__device__ __forceinline__ void wait_async_lds() {
#if __has_builtin(__builtin_amdgcn_s_wait_asynccnt)
  __builtin_amdgcn_s_wait_asynccnt(0);
#else
  asm volatile("s_wait_asynccnt 0" ::: "memory");
#endif
}

// ---------------- CSC build ----------------
__global__ void k_zero_i32(int* __restrict__ p, int n) {
  int i = blockIdx.x * blockDim.x + threadIdx.x;
  if (i < n) p[i] = 0;
}

__global__ void k_hist(const int* __restrict__ cols, int* __restrict__ hist, int nnz) {
  int k = blockIdx.x * blockDim.x + threadIdx.x;
  if (k < nnz) atomicAdd(&hist[cols[k]], 1);
}

// single-block exclusive scan over n (<= 16384) bins; writes offs[0..n] and cursor[0..n-1]
__global__ __launch_bounds__(1024) void k_scan(const int* __restrict__ hist,
                                               int* __restrict__ offs,
                                               int* __restrict__ cursor, int n) {
  __shared__ int sdata[1024];
  const int T = 1024;
  int tid = threadIdx.x;
  int items = (n + T - 1) / T;  // 8 for n=8192
  if (items > 16) items = 16;
  int loc[16];
  int base = tid * items;
  int tsum = 0;
  for (int i = 0; i < items; ++i) {
    int idx = base + i;
    int e = (idx < n) ? hist[idx] : 0;
    loc[i] = tsum;
    tsum += e;
  }
  sdata[tid] = tsum;
  __syncthreads();
  for (int d = 1; d < T; d <<= 1) {           // Hillis-Steele inclusive scan
    int v = (tid >= d) ? sdata[tid - d] : 0;
    __syncthreads();
    sdata[tid] += v;
    __syncthreads();
  }
  int excl = sdata[tid] - tsum;
  for (int i = 0; i < items; ++i) {
    int idx = base + i;
    if (idx < n) {
      int o = excl + loc[i];
      offs[idx] = o;
      cursor[idx] = o;
    }
  }
  if (tid == T - 1) offs[n] = sdata[T - 1];
}

__global__ void k_fill(const int* __restrict__ rows, const int* __restrict__ cols,
                       const float* __restrict__ vals, int* __restrict__ cursor,
                       int* __restrict__ cscRow, float* __restrict__ cscVal, int nnz) {
  int k = blockIdx.x * blockDim.x + threadIdx.x;
  if (k < nnz) {
    int c = cols[k];
    int p = atomicAdd(&cursor[c], 1);
    cscRow[p] = rows[k];
    cscVal[p] = vals[k];
  }
}

// ---------------- transpose via WMMA identity trick ----------------
// dst[c*sr + r] = src[r*sc + c].
// One wave per 16x16 tile. D = sum_kb A_kb x I_kb where A_kb[m][k] = T[kb+k][m]
// (A-operand layout: lane = M, VGPRs/lane-halves = K) and I_kb is the identity
// slice in B-operand layout. Result D = T^T lands in C/D layout (row per VGPR,
// lane = N) and is stored with coalesced 64B half-wave segments. No LDS, no
// barriers; matrix pipe does the lane shuffle.
__global__ __launch_bounds__(256) void k_transpose_wmma(const float* __restrict__ src,
                                                        float* __restrict__ dst,
                                                        int sr, int sc) {
  int tilesC = sc >> 4;
  int tilesR = (sr + 15) >> 4;
  int wave = blockIdx.x * (blockDim.x >> 5) + (threadIdx.x >> 5);
  if (wave >= tilesC * tilesR) return;              // wave-uniform exit
  int tc = wave % tilesC, tr = wave / tilesC;
  int r0 = tr << 4, c0 = tc << 4;
  int lane = threadIdx.x & 31;
  int hi = lane >> 4;       // 0: lanes 0-15, 1: lanes 16-31
  int lm = lane & 15;

  if (r0 + 16 <= sr && c0 + 16 <= sc) {             // full tile (wave-uniform)
    v8f acc = {};
#pragma unroll
    for (int kb = 0; kb < 16; kb += 4) {
      v2f a, b;
      // A-operand: lane (hi,lm) holds A[m=lm][k=2*hi+j] = T[kb+2*hi+j][lm]
      a.x = src[(size_t)(r0 + kb + 2 * hi) * sc + (c0 + lm)];
      a.y = src[(size_t)(r0 + kb + 2 * hi + 1) * sc + (c0 + lm)];
      // B-operand: identity slice, B[k][n] = (kb+k == n)
      b.x = (lm == kb + 2 * hi) ? 1.0f : 0.0f;
      b.y = (lm == kb + 2 * hi + 1) ? 1.0f : 0.0f;
      acc = __builtin_amdgcn_wmma_f32_16x16x4_f32(false, a, false, b,
                                                  (short)0, acc, false, false);
    }
    // D layout: VGPR v -> row m = v + 8*hi, col n = lm ; D[m][n] = T[n][m]
#pragma unroll
    for (int v = 0; v < 8; ++v) {
      dst[(size_t)(c0 + v + 8 * hi) * sr + (r0 + lm)] = acc[v];
    }
  } else {                                          // partial tile fallback
    for (int i = lane; i < 256; i += 32) {
      int rr = r0 + (i >> 4), cc = c0 + (i & 15);
      if (rr < sr && cc < sc) dst[(size_t)cc * sr + rr] = src[(size_t)rr * sc + cc];
    }
  }
}

// ---------------- hot path: one SpMM + bias + relu step ----------------
// cur, nxt are N x B (transposed layout). Block = one output column c,
// 256 threads * float4 = B=1024 lanes of batch.
#define META_CHUNK 128

__device__ __forceinline__ float4 spmm_col_acc(const float* __restrict__ cur,
                                               const int* __restrict__ offs,
                                               const int* __restrict__ cscRow,
                                               const float* __restrict__ cscVal,
                                               int c, int tid, int Bq,
                                               int* s_row, float* s_val) {
  int j0 = offs[c], j1 = offs[c + 1];
  const float4* cur4 = (const float4*)cur;
  float4 acc = make_float4(0.f, 0.f, 0.f, 0.f);
  for (int jb = j0; jb < j1; jb += META_CHUNK) {
    int cnt = min(META_CHUNK, j1 - jb);
    if (tid < cnt) {  // async-stage this chunk's CSC metadata into LDS
      async_g2l_b32(cscRow + jb + tid, &s_row[tid]);
      async_g2l_b32(cscVal + jb + tid, &s_val[tid]);
    }
    wait_async_lds();
    __syncthreads();
    for (int t = 0; t < cnt; ++t) {
      int r = s_row[t];
      float v = s_val[t];
      float4 s = cur4[(size_t)r * Bq + tid];
      acc.x += v * s.x;
      acc.y += v * s.y;
      acc.z += v * s.z;
      acc.w += v * s.w;
    }
    __syncthreads();
  }
  return acc;
}

__global__ __launch_bounds__(256) void k_spmm_step(const float* __restrict__ cur,
                                                   float* __restrict__ nxt,
                                                   const float* __restrict__ bias,
                                                   const int* __restrict__ offs,
                                                   const int* __restrict__ cscRow,
                                                   const float* __restrict__ cscVal,
                                                   int Bq) {
  __shared__ int s_row[META_CHUNK];
  __shared__ float s_val[META_CHUNK];
  int c = blockIdx.x;
  int tid = threadIdx.x;
  float4 acc = spmm_col_acc(cur, offs, cscRow, cscVal, c, tid, Bq, s_row, s_val);
  float bc = bias[c];
  acc.x = fmaxf(acc.x + bc, 0.f);
  acc.y = fmaxf(acc.y + bc, 0.f);
  acc.z = fmaxf(acc.z + bc, 0.f);
  acc.w = fmaxf(acc.w + bc, 0.f);
  ((float4*)nxt)[(size_t)c * Bq + tid] = acc;
}

// final-step variant writing directly in B x N layout (used only if ws is small)
__global__ __launch_bounds__(256) void k_spmm_step_scatter(const float* __restrict__ cur,
                                                           float* __restrict__ out,
                                                           const float* __restrict__ bias,
                                                           const int* __restrict__ offs,
                                                           const int* __restrict__ cscRow,
                                                           const float* __restrict__ cscVal,
                                                           int Bq, int N) {
  __shared__ int s_row[META_CHUNK];
  __shared__ float s_val[META_CHUNK];
  int c = blockIdx.x;
  int tid = threadIdx.x;
  float4 acc = spmm_col_acc(cur, offs, cscRow, cscVal, c, tid, Bq, s_row, s_val);
  float bc = bias[c];
  int b0 = tid * 4;
  out[(size_t)(b0 + 0) * N + c] = fmaxf(acc.x + bc, 0.f);
  out[(size_t)(b0 + 1) * N + c] = fmaxf(acc.y + bc, 0.f);
  out[(size_t)(b0 + 2) * N + c] = fmaxf(acc.z + bc, 0.f);
  out[(size_t)(b0 + 3) * N + c] = fmaxf(acc.w + bc, 0.f);
}

extern "C" void kernel_launch(void* const* d_in, const int* in_sizes, int n_in,
                              void* d_out, int out_size, void* d_ws, size_t ws_size,
                              hipStream_t stream) {
  const float* x = (const float*)d_in[0];
  const float* vals = (const float*)d_in[1];
  const float* bias = (const float*)d_in[2];
  const int* rows = (const int*)d_in[3];
  const int* cols = (const int*)d_in[4];

  const int BN = in_sizes[0];   // B*N
  const int NNZ = in_sizes[1];
  const int N = in_sizes[2];    // bias length
  const int B = BN / N;
  const int Bq = B / 4;

  char* ws = (char*)d_ws;
  size_t off = 0;
  float* bufA = (float*)(ws + off); off += (size_t)BN * 4;
  int* hist   = (int*)(ws + off);   off += (size_t)N * 4;
  int* offs   = (int*)(ws + off);   off += (size_t)(N + 1) * 4;
  int* cursor = (int*)(ws + off);   off += (size_t)N * 4;
  int* cscRow = (int*)(ws + off);   off += (size_t)NNZ * 4;
  float* cscVal = (float*)(ws + off); off += (size_t)NNZ * 4;
  off = (off + 255) & ~(size_t)255;
  const bool two_buf = (ws_size >= off + (size_t)BN * 4);
  float* bufB = two_buf ? (float*)(ws + off) : nullptr;
  float* outF = (float*)d_out;

  // ---- build CSC (amortized over 3 steps) ----
  k_zero_i32<<<(N + 255) / 256, 256, 0, stream>>>(hist, N);
  k_hist<<<(NNZ + 255) / 256, 256, 0, stream>>>(cols, hist, NNZ);
  k_scan<<<1, 1024, 0, stream>>>(hist, offs, cursor, N);
  k_fill<<<(NNZ + 255) / 256, 256, 0, stream>>>(rows, cols, vals, cursor, cscRow, cscVal, NNZ);

  // transpose launches: one wave per 16x16 tile, 8 waves per block
  auto tr_grid = [](int sr, int sc) {
    long tiles = (long)((sr + 15) / 16) * (sc / 16);
    return (int)((tiles + 7) / 8);
  };

  // x (B x N) -> bufA (N x B)
  k_transpose_wmma<<<tr_grid(B, N), 256, 0, stream>>>(x, bufA, B, N);

  if (two_buf) {
    k_spmm_step<<<N, 256, 0, stream>>>(bufA, bufB, bias, offs, cscRow, cscVal, Bq);
    k_spmm_step<<<N, 256, 0, stream>>>(bufB, bufA, bias, offs, cscRow, cscVal, Bq);
    k_spmm_step<<<N, 256, 0, stream>>>(bufA, bufB, bias, offs, cscRow, cscVal, Bq);
    // bufB (N x B) -> d_out (B x N)
    k_transpose_wmma<<<tr_grid(N, B), 256, 0, stream>>>(bufB, outF, N, B);
  } else {
    // use d_out as N x B scratch for the middle step
    k_spmm_step<<<N, 256, 0, stream>>>(bufA, outF, bias, offs, cscRow, cscVal, Bq);
    k_spmm_step<<<N, 256, 0, stream>>>(outF, bufA, bias, offs, cscRow, cscVal, Bq);
    k_spmm_step_scatter<<<N, 256, 0, stream>>>(bufA, outF, bias, offs, cscRow, cscVal, Bq, N);
  }
}